// GGNN_87917980549370
// MI455X (gfx1250) — compile-verified
//
#include <hip/hip_runtime.h>
#include <hip/hip_bf16.h>
#include <math.h>

#define NN 100000
#define EE 1000000
#define HH 64
#define MM 64
#define CC 16
#define GG 192   // 3*H gate width

typedef __attribute__((ext_vector_type(2))) float v2f;
typedef __attribute__((ext_vector_type(8))) float v8f;

// D = A(16x4 f32) * B(4x16 f32) + C(16x16 f32), full fp32 precision on the matrix pipe
static __device__ __forceinline__ v8f wmma_f32(v2f a, v2f b, v8f c) {
  return __builtin_amdgcn_wmma_f32_16x16x4_f32(false, a, false, b, (short)0, c,
                                               false, false);
}

// Load an A or B fragment for 16x16x4 f32 WMMA from a row-major matrix
// (works for global or LDS pointers).
// A: rows = M (lane r), VGPR/half = K.  B: rows of W are the N dim (lane r = col),
// contraction dim K along the row -> identical load pattern.
static __device__ __forceinline__ v2f frag_rm(const float* p, int row0, int ld,
                                              int k0, int r, int half) {
  return *(const v2f*)(p + (size_t)(row0 + r) * ld + k0 + 2 * half);
}

// ---------------------------------------------------------------- zero scratch
__global__ void k_zero(float* __restrict__ p, long n) {
  long i = (long)blockIdx.x * blockDim.x + threadIdx.x;
  long stride = (long)gridDim.x * blockDim.x;
  for (; i < n; i += stride) p[i] = 0.0f;
}

// ----------------------------------------------- fused weights Wf = W_ih @ W{0,1}
// Wf0[oc,k] = sum_m W_ih[oc,m] * W0[m,k];  vb0[oc] = sum_m W_ih[oc,m] * b0[m]
__global__ void k_fuse(const float* __restrict__ W_ih, const float* __restrict__ W0,
                       const float* __restrict__ b0, const float* __restrict__ W1,
                       const float* __restrict__ b1, float* __restrict__ Wf0,
                       float* __restrict__ Wf1, float* __restrict__ vb0,
                       float* __restrict__ vb1) {
  const int WN = GG * MM;  // 12288
  int idx = blockIdx.x * blockDim.x + threadIdx.x;
  if (idx < WN) {
    int oc = idx / MM, k = idx % MM;
    float s = 0.f;
    for (int m = 0; m < MM; ++m) s += W_ih[oc * MM + m] * W0[m * HH + k];
    Wf0[idx] = s;
  } else if (idx < 2 * WN) {
    int j = idx - WN, oc = j / MM, k = j % MM;
    float s = 0.f;
    for (int m = 0; m < MM; ++m) s += W_ih[oc * MM + m] * W1[m * HH + k];
    Wf1[j] = s;
  } else if (idx < 2 * WN + GG) {
    int oc = idx - 2 * WN;
    float s = 0.f;
    for (int m = 0; m < MM; ++m) s += W_ih[oc * MM + m] * b0[m];
    vb0[oc] = s;
  } else if (idx < 2 * WN + 2 * GG) {
    int oc = idx - 2 * WN - GG;
    float s = 0.f;
    for (int m = 0; m < MM; ++m) s += W_ih[oc * MM + m] * b1[m];
    vb1[oc] = s;
  }
}

// ------------------------------------- per-edge gather + f32 atomic scatter-add
// 16 threads per edge, each handles 4 contiguous columns (float4 gather).
// edge_type True -> slot 0 (W0/b0 path), False -> slot 1.
__global__ void k_edges(const float* __restrict__ feat, const int* __restrict__ src,
                        const int* __restrict__ dst,
                        const unsigned char* __restrict__ et,
                        float* __restrict__ aggX, float* __restrict__ cnt) {
  int tid = blockIdx.x * blockDim.x + threadIdx.x;
  int e = tid >> 4;
  if (e >= EE) return;
  int c4 = (tid & 15) * 4;
  int s = src[e];
  int d = dst[e];
  int t = (et[e] != 0) ? 0 : 1;
  const float4 x = *(const float4*)(feat + (size_t)s * HH + c4);
  float* a = aggX + ((size_t)t * NN + (size_t)d) * HH + c4;
  unsafeAtomicAdd(a + 0, x.x);
  unsafeAtomicAdd(a + 1, x.y);
  unsafeAtomicAdd(a + 2, x.z);
  unsafeAtomicAdd(a + 3, x.w);
  if ((tid & 15) == 0) unsafeAtomicAdd(cnt + (size_t)t * NN + d, 1.0f);
}

// -------------------------------------------------- fused GRU + classifier WMMA
// One 16-node tile per wave. All gate weights (144 KB) staged once per block in
// LDS (WGP has 320 KB; WGP$ is only 64 KB so global re-reads would miss).
// Accumulators:
//   c_rz[8] : gates r,z cols 0..127 of gi+gh (summed since r=sigmoid(i_r+h_r))
//   c_in[4] : i_n cols 128..191 (agg side), c_hn[4] : h_n cols 128..191 (hidden)
// Dynamic LDS layout (floats):
//   [0)       Wf0  12288
//   [12288)   Wf1  12288
//   [24576)   W_hh 12288
//   [36864)   h tiles: 8 waves x 16 rows x 68 (padded)
#define SM_WF0 0
#define SM_WF1 12288
#define SM_WHH 24576
#define SM_H   36864
#define SM_FLOATS (36864 + 8 * 16 * 68)

__global__ void __launch_bounds__(256) k_gru(
    const float* __restrict__ feat, const float* __restrict__ aggX,
    const float* __restrict__ cnt, const float* __restrict__ Wf0,
    const float* __restrict__ W_hh, const float* __restrict__ vb0,
    const float* __restrict__ vb1, const float* __restrict__ b_ih,
    const float* __restrict__ b_hh, const float* __restrict__ W_out,
    const float* __restrict__ b_out, float* __restrict__ out) {
  extern __shared__ float smem[];

  const int lane = threadIdx.x & 31;
  const int wv = threadIdx.x >> 5;
  const int r = lane & 15;
  const int half = lane >> 4;

  // ---- stage Wf0|Wf1 (contiguous in ws) and W_hh into LDS, float4 strides ----
  {
    const float4* wf4 = (const float4*)Wf0;     // 24576 floats = 6144 float4
    const float4* wh4 = (const float4*)W_hh;    // 12288 floats = 3072 float4
    float4* s4 = (float4*)smem;
    for (int i = threadIdx.x; i < 6144; i += 256) s4[i] = wf4[i];
    for (int i = threadIdx.x; i < 3072; i += 256) s4[6144 + i] = wh4[i];
  }
  __syncthreads();

  const float* sWf0 = smem + SM_WF0;
  const float* sWf1 = smem + SM_WF1;
  const float* sWhh = smem + SM_WHH;
  float* hrow = smem + SM_H + wv * (16 * 68);

  int tile = blockIdx.x * 8 + wv;
  const int ntiles = NN / 16;             // 6250, N divisible by 16
  if (tile >= ntiles) tile = ntiles - 1;  // clamp: keep EXEC all-ones (benign dup)
  const int n0 = tile * 16;

  const float* agg0 = aggX;
  const float* agg1 = aggX + (size_t)NN * HH;

  v8f c_rz[8], c_in[4], c_hn[4];
  const v8f z8 = {0.f, 0.f, 0.f, 0.f, 0.f, 0.f, 0.f, 0.f};
#pragma unroll
  for (int t = 0; t < 8; ++t) c_rz[t] = z8;
#pragma unroll
  for (int t = 0; t < 4; ++t) { c_in[t] = z8; c_hn[t] = z8; }

  for (int k0 = 0; k0 < HH; k0 += 4) {
    v2f aA0 = frag_rm(agg0, n0, HH, k0, r, half);
    v2f aA1 = frag_rm(agg1, n0, HH, k0, r, half);
    v2f aF  = frag_rm(feat, n0, HH, k0, r, half);
#pragma unroll
    for (int t = 0; t < 8; ++t) {
      c_rz[t] = wmma_f32(aA0, frag_rm(sWf0, t * 16, MM, k0, r, half), c_rz[t]);
      c_rz[t] = wmma_f32(aA1, frag_rm(sWf1, t * 16, MM, k0, r, half), c_rz[t]);
      c_rz[t] = wmma_f32(aF, frag_rm(sWhh, t * 16, HH, k0, r, half), c_rz[t]);
    }
#pragma unroll
    for (int t = 0; t < 4; ++t) {
      c_in[t] = wmma_f32(aA0, frag_rm(sWf0, 128 + t * 16, MM, k0, r, half), c_in[t]);
      c_in[t] = wmma_f32(aA1, frag_rm(sWf1, 128 + t * 16, MM, k0, r, half), c_in[t]);
      c_hn[t] = wmma_f32(aF, frag_rm(sWhh, 128 + t * 16, HH, k0, r, half), c_hn[t]);
    }
  }

  float cnt0v[8], cnt1v[8];
#pragma unroll
  for (int i = 0; i < 8; ++i) {
    int node = n0 + half * 8 + i;
    cnt0v[i] = cnt[node];
    cnt1v[i] = cnt[NN + node];
  }

#pragma unroll
  for (int t = 0; t < 4; ++t) {
    const int cr = t * 16 + r;   // r-gate / feature / n-gate column within H
    const int cz = 64 + cr;
    const int cn = 128 + cr;
    const float br = b_ih[cr] + b_hh[cr];
    const float bz = b_ih[cz] + b_hh[cz];
    const float vbr0 = vb0[cr], vbr1 = vb1[cr];
    const float vbz0 = vb0[cz], vbz1 = vb1[cz];
    const float bin = b_ih[cn], vbn0 = vb0[cn], vbn1 = vb1[cn];
    const float bhn = b_hh[cn];
#pragma unroll
    for (int i = 0; i < 8; ++i) {
      const int node = n0 + half * 8 + i;
      float rg = 1.0f / (1.0f + expf(-(c_rz[t][i] + br + cnt0v[i] * vbr0 + cnt1v[i] * vbr1)));
      float zg = 1.0f / (1.0f + expf(-(c_rz[t + 4][i] + bz + cnt0v[i] * vbz0 + cnt1v[i] * vbz1)));
      float inv = c_in[t][i] + bin + cnt0v[i] * vbn0 + cnt1v[i] * vbn1;
      float hnv = c_hn[t][i] + bhn;
      float ng = tanhf(inv + rg * hnv);
      float fv = feat[(size_t)node * HH + cr];
      float hv = (1.0f - zg) * ng + zg * fv;
      hrow[(half * 8 + i) * 68 + cr] = hv;  // D-layout -> row-major in LDS
    }
  }
  __syncthreads();  // uniform: all waves active (clamped tiles)

  // classifier: out = h @ W_out^T + b_out, h re-read from LDS in A-layout
  v8f co = z8;
  for (int k0 = 0; k0 < HH; k0 += 4) {
    v2f ah = *(const v2f*)(hrow + r * 68 + k0 + 2 * half);
    v2f bo = frag_rm(W_out, 0, HH, k0, r, half);
    co = wmma_f32(ah, bo, co);
  }
  const float bov = b_out[r];
#pragma unroll
  for (int i = 0; i < 8; ++i) {
    int node = n0 + half * 8 + i;
    out[(size_t)node * CC + r] = co[i] + bov;
  }
}

extern "C" void kernel_launch(void* const* d_in, const int* in_sizes, int n_in,
                              void* d_out, int out_size, void* d_ws, size_t ws_size,
                              hipStream_t stream) {
  const float* feat = (const float*)d_in[0];
  const int* src = (const int*)d_in[1];
  const int* dst = (const int*)d_in[2];
  const unsigned char* et = (const unsigned char*)d_in[3];  // jax bool = 1 byte
  const float* W0 = (const float*)d_in[4];
  const float* b0 = (const float*)d_in[5];
  const float* W1 = (const float*)d_in[6];
  const float* b1 = (const float*)d_in[7];
  const float* W_ih = (const float*)d_in[8];
  const float* W_hh = (const float*)d_in[9];
  const float* b_ih = (const float*)d_in[10];
  const float* b_hh = (const float*)d_in[11];
  const float* W_out = (const float*)d_in[12];
  const float* b_out = (const float*)d_in[13];

  // workspace layout (floats): aggX[2][N][64], cnt[2][N], Wf0/Wf1[192][64], vb0/vb1[192]
  float* ws = (float*)d_ws;
  float* aggX = ws;
  float* cnt = aggX + (size_t)2 * NN * HH;
  float* Wf0 = cnt + (size_t)2 * NN;   // Wf1 contiguous after Wf0
  float* Wf1 = Wf0 + GG * MM;
  float* vb0 = Wf1 + GG * MM;
  float* vb1 = vb0 + GG;

  long nz = (long)2 * NN * HH + 2 * NN;
  k_zero<<<2048, 256, 0, stream>>>(aggX, nz);
  k_fuse<<<(2 * GG * MM + 2 * GG + 255) / 256, 256, 0, stream>>>(
      W_ih, W0, b0, W1, b1, Wf0, Wf1, vb0, vb1);
  k_edges<<<(EE * 16 + 255) / 256, 256, 0, stream>>>(feat, src, dst, et, aggX, cnt);
  const int tiles = NN / 16;
  const size_t smem_bytes = (size_t)SM_FLOATS * sizeof(float);  // ~182 KB < 320 KB
  k_gru<<<(tiles + 7) / 8, 256, smem_bytes, stream>>>(
      feat, aggX, cnt, Wf0, W_hh, vb0, vb1, b_ih, b_hh, W_out, b_out,
      (float*)d_out);
}